// Slice_2362232013449
// MI455X (gfx1250) — compile-verified
//
#include <hip/hip_runtime.h>
#include <stdint.h>

// Problem constants (N, C, D, H, W) = (4, 9, 16, 1024, 1024)
#define NIMG 4
#define NCH 9
#define DD 16
#define HWPX (1024 * 1024)
#define GRID_ELEMS (NCH * DD * DD * DD)   // 36864 floats
#define LDS_BYTES (GRID_ELEMS * 4)        // 147456 B -> 2 blocks / WGP (320KB)
#define BLOCK 256
#define PX_PER_BLOCK 4096                 // 16 px / thread
#define BLOCKS_PER_IMG (HWPX / PX_PER_BLOCK)  // 256

#ifndef USE_TDM
#if __has_builtin(__builtin_amdgcn_tensor_load_to_lds)
#define USE_TDM 1
#else
#define USE_TDM 0
#endif
#endif

typedef __attribute__((ext_vector_type(4))) unsigned int u32x4;
typedef __attribute__((ext_vector_type(8))) int i32x8;
typedef __attribute__((ext_vector_type(4))) int i32x4;

// One pixel: trilinear slice of 9 channels from LDS grid + fused 3x12 conv.
__device__ __forceinline__ void one_px(const float* __restrict__ s,
                                       const float* __restrict__ w,
                                       float cb0, float cb1, float cb2,
                                       float r, float g, float b,
                                       float& y0, float& y1, float& y2) {
  float crd = fminf(fmaxf(r, 0.f), 1.f) * 15.f;
  float cgd = fminf(fmaxf(g, 0.f), 1.f) * 15.f;
  float cbd = fminf(fmaxf(b, 0.f), 1.f) * 15.f;
  int ir = (int)crd; ir = ir > 14 ? 14 : ir;
  int ig = (int)cgd; ig = ig > 14 ? 14 : ig;
  int ib = (int)cbd; ib = ib > 14 ? 14 : ib;
  float fr = crd - (float)ir;
  float fg = cgd - (float)ig;
  float fb = cbd - (float)ib;
  const float* p = s + ((ir * 16 + ig) * 16 + ib);

  // guide passthrough channels (concat) folded into the conv
  float a0 = fmaf(w[9],  r, fmaf(w[10], g, fmaf(w[11], b, cb0)));
  float a1 = fmaf(w[21], r, fmaf(w[22], g, fmaf(w[23], b, cb1)));
  float a2 = fmaf(w[33], r, fmaf(w[34], g, fmaf(w[35], b, cb2)));

#pragma unroll
  for (int c = 0; c < 9; ++c) {
    const float* q = p + c * 4096;  // channel stride in grid
    // b-neighbors are adjacent floats -> ds_load_2addr_b32 pairs
    float v000 = q[0],   v001 = q[1];
    float v010 = q[16],  v011 = q[17];
    float v100 = q[256], v101 = q[257];
    float v110 = q[272], v111 = q[273];
    float b00 = fmaf(fb, v001 - v000, v000);
    float b01 = fmaf(fb, v011 - v010, v010);
    float b10 = fmaf(fb, v101 - v100, v100);
    float b11 = fmaf(fb, v111 - v110, v110);
    float g0v = fmaf(fg, b01 - b00, b00);
    float g1v = fmaf(fg, b11 - b10, b10);
    float v   = fmaf(fr, g1v - g0v, g0v);
    a0 = fmaf(w[c],      v, a0);
    a1 = fmaf(w[12 + c], v, a1);
    a2 = fmaf(w[24 + c], v, a2);
  }
  y0 = a0; y1 = a1; y2 = a2;
}

__global__ __launch_bounds__(BLOCK)
void bgrid_slice_kernel(const float* __restrict__ grid,
                        const float* __restrict__ guide,
                        const float* __restrict__ cw,
                        const float* __restrict__ cb,
                        float* __restrict__ out) {
  extern __shared__ float smem[];
  const int n = blockIdx.y;
  const float* gsrc = grid + (size_t)n * GRID_ELEMS;

#if USE_TDM
  // Tensor Data Mover: DMA the whole 144KB grid[n] into LDS (wave 0 issues).
  if (threadIdx.x < 32) {
    uint64_t ga = (uint64_t)(uintptr_t)gsrc;
    uint32_t lds = (uint32_t)(uintptr_t)smem;  // flat LDS addr: low 32 bits = offset
    // D# group0: count=1 | lds_addr | global_addr[56:0] | type=2
    u32x4 g0 = { 1u, lds, (uint32_t)ga,
                 (uint32_t)(ga >> 32) | 0x80000000u };
    // D# group1: 2D tensor 4096 x 9, tile 4096 x 9, dword elements,
    // stride0 = 4096 elems; no multicast/pad/iterate/atomic-barrier.
    i32x8 g1 = { (int)0x00020000u,   // [17:16] data_size = 2 (4 bytes)
                 (int)0x10000000u,   // [63:48] tensor_dim0.lo = 4096
                 (int)0x00090000u,   // [95:80] tensor_dim1.lo = 9
                 (int)0x10000000u,   // [127:112] tile_dim0 = 4096
                 9,                  // [143:128] tile_dim1 = 9 (tile_dim2 = 0)
                 4096,               // [191:160] tensor_dim0_stride.lo
                 0, 0 };
    i32x4 gz4 = { 0, 0, 0, 0 };      // groups 2/3 unused (2D tensor)
    i32x8 gz8 = { 0, 0, 0, 0, 0, 0, 0, 0 };
    __builtin_amdgcn_tensor_load_to_lds(g0, g1, gz4, gz4, gz8, 0);
    __builtin_amdgcn_s_wait_tensorcnt(0);
  }
  __syncthreads();
#else
  {
    const float4* s4 = (const float4*)gsrc;
    float4* d4 = (float4*)smem;
    for (int i = threadIdx.x; i < GRID_ELEMS / 4; i += BLOCK) d4[i] = s4[i];
    __syncthreads();
  }
#endif

  // Conv params are uniform -> SGPRs.
  float w[36];
#pragma unroll
  for (int i = 0; i < 36; ++i) w[i] = cw[i];
  float cb0 = cb[0], cb1 = cb[1], cb2 = cb[2];

  const float* gb = guide + (size_t)n * 3 * HWPX;
  float*       ob = out   + (size_t)n * 3 * HWPX;
  int p = blockIdx.x * PX_PER_BLOCK + threadIdx.x * 4;

#pragma unroll
  for (int it = 0; it < PX_PER_BLOCK / (BLOCK * 4); ++it, p += BLOCK * 4) {
    if (it + 1 < PX_PER_BLOCK / (BLOCK * 4)) {
      __builtin_prefetch(gb + p + BLOCK * 4, 0, 0);              // global_prefetch
      __builtin_prefetch(gb + HWPX + p + BLOCK * 4, 0, 0);
      __builtin_prefetch(gb + 2 * HWPX + p + BLOCK * 4, 0, 0);
    }
    float4 xr = *(const float4*)(gb + p);
    float4 xg = *(const float4*)(gb + HWPX + p);
    float4 xb = *(const float4*)(gb + 2 * HWPX + p);
    float4 o0, o1, o2;
    const float* rr = (const float*)&xr;
    const float* gg = (const float*)&xg;
    const float* bb = (const float*)&xb;
    float* q0 = (float*)&o0;
    float* q1 = (float*)&o1;
    float* q2 = (float*)&o2;
#pragma unroll
    for (int k = 0; k < 4; ++k) {
      one_px(smem, w, cb0, cb1, cb2, rr[k], gg[k], bb[k], q0[k], q1[k], q2[k]);
    }
    *(float4*)(ob + p)            = o0;
    *(float4*)(ob + HWPX + p)     = o1;
    *(float4*)(ob + 2 * HWPX + p) = o2;
  }
}

extern "C" void kernel_launch(void* const* d_in, const int* in_sizes, int n_in,
                              void* d_out, int out_size, void* d_ws, size_t ws_size,
                              hipStream_t stream) {
  const float* grid  = (const float*)d_in[0];  // (4, 9, 16, 16, 16)
  const float* guide = (const float*)d_in[1];  // (4, 3, 1024, 1024)
  const float* cw    = (const float*)d_in[2];  // (3, 12)
  const float* cbv   = (const float*)d_in[3];  // (3,)
  float* out = (float*)d_out;                  // (4, 3, 1024, 1024)

  (void)in_sizes; (void)n_in; (void)out_size; (void)d_ws; (void)ws_size;

  // 144KB dynamic LDS (> default cap) — deterministic, capture-safe.
  (void)hipFuncSetAttribute((const void*)bgrid_slice_kernel,
                            hipFuncAttributeMaxDynamicSharedMemorySize, LDS_BYTES);

  dim3 grd(BLOCKS_PER_IMG, NIMG);
  bgrid_slice_kernel<<<grd, BLOCK, LDS_BYTES, stream>>>(grid, guide, cw, cbv, out);
}